// pca_90348932039345
// MI455X (gfx1250) — compile-verified
//
#include <hip/hip_runtime.h>
#include <hip/hip_bf16.h>

// CDNA5 fp32 WMMA fragments
typedef __attribute__((ext_vector_type(2))) float v2f;
typedef __attribute__((ext_vector_type(8))) float v8f;
// TDM descriptor groups
typedef __attribute__((ext_vector_type(4))) unsigned int u32x4;
typedef __attribute__((ext_vector_type(8))) int i32x8;
typedef __attribute__((ext_vector_type(4))) int i32x4;

#if defined(__has_builtin)
#if __has_builtin(__builtin_amdgcn_tensor_load_to_lds)
#define HAVE_TDM 1
#endif
#endif
#if __has_include(<hip/amd_detail/amd_gfx1250_TDM.h>)
#define TDM_SIX_ARG 1   // therock-10.0 headers -> 6-arg builtin arity
#endif

#define LDA 65   // 64 + 1 pad -> conflict-free column walks in 64-bank LDS
#define LDV 17
#define NPC 4
#define NITER 16

// A-fragment: 16x4 f32 (MxK). Lanes 0-15: K=0,1 ; lanes 16-31: K=2,3. (ISA 7.12.2)
__device__ __forceinline__ v2f ldA(const float* S, int row0, int k0, int lane) {
    const int half = lane >> 4;
    const int m = row0 + (lane & 15);
    v2f a;
    a.x = S[m * LDA + k0 + 2 * half + 0];
    a.y = S[m * LDA + k0 + 2 * half + 1];
    return a;
}
// B-fragment: 4x16 f32 (KxN). Lanes 0-15: rows K=0,1 ; lanes 16-31: rows K=2,3.
__device__ __forceinline__ v2f ldB(const float* S, int ldb, int k0, int col0, int lane) {
    const int half = lane >> 4;
    const int n = col0 + (lane & 15);
    v2f b;
    b.x = S[(k0 + 2 * half + 0) * ldb + n];
    b.y = S[(k0 + 2 * half + 1) * ldb + n];
    return b;
}
__device__ __forceinline__ v8f wmma4(v2f a, v2f b, v8f c) {
    return __builtin_amdgcn_wmma_f32_16x16x4_f32(false, a, false, b, (short)0, c,
                                                 false, false);
}
// One 16x16 tile over K=64 (16 wmma)
__device__ __forceinline__ v8f tile64(const float* SA, const float* SB, int ldb,
                                      int row0, int col0, int lane) {
    v8f c = {};
#pragma unroll
    for (int k0 = 0; k0 < 64; k0 += 4)
        c = wmma4(ldA(SA, row0, k0, lane), ldB(SB, ldb, k0, col0, lane), c);
    return c;
}
// 16x64 row-block (4 tiles), A-fragment hoisted across the 4 N-tiles
__device__ __forceinline__ void rowblock64(const float* SA, const float* SB, int ldb,
                                           int row0, int lane, v8f acc[4]) {
#pragma unroll
    for (int tn = 0; tn < 4; ++tn) acc[tn] = v8f{};
#pragma unroll
    for (int k0 = 0; k0 < 64; k0 += 4) {
        v2f a = ldA(SA, row0, k0, lane);
#pragma unroll
        for (int tn = 0; tn < 4; ++tn)
            acc[tn] = wmma4(a, ldB(SB, ldb, k0, tn * 16, lane), acc[tn]);
    }
}
// wave32 butterfly sum
__device__ __forceinline__ float waveRed(float v) {
#pragma unroll
    for (int off = 16; off > 0; off >>= 1) v += __shfl_xor(v, off, 32);
    return v;
}

__global__ __launch_bounds__(128)
void pca_wmma_kernel(const float* __restrict__ X, float* __restrict__ out) {
    __shared__ float sA [64 * LDA];  // A (64x64), TDM-padded stride 65
    __shared__ float sC [64 * LDA];  // cov, later G = cov*cov^T
    __shared__ float sCT[64 * LDA];  // cov^T
    __shared__ float sV [64 * LDV];  // V padded 64x16 (cols 4..15 = 0)
    __shared__ float sW [64 * LDV];  // W = G*V
    __shared__ float sPart[4];

    const int b    = blockIdx.x;     // 0..8191
    const int mi   = b >> 7;         // n = 128
    const int ni   = b & 127;
    const int tid  = threadIdx.x;
    const int lane = tid & 31;
    const int wave = tid >> 5;       // 4 waves
    const int row0 = wave * 16;

    const float* Xb = X + (size_t)mi * 64 * 64 * 128 + ni;

#if HAVE_TDM
    // ---- Tensor Data Mover: gather A[p][q] = X[mi,p,q,ni] (elem stride 512B,
    //      row stride 32KB) into LDS with hardware padding (64 DWORDs data,
    //      1 DWORD pad -> stride-65 layout). D# per ISA ch.8, 3D view:
    //      tile_dim0=1, tile_dim1=64 (stride 128), tile_dim2=64 (stride 8192).
    if (wave == 0) {
        const unsigned long long ga = (unsigned long long)(uintptr_t)Xb;  // byte addr
        const unsigned int la = (unsigned int)(uintptr_t)&sA[0];          // LDS offset
        u32x4 g0 = { 1u,                                   // count=1 (valid user D#)
                     la,                                   // lds_addr
                     (unsigned int)ga,                     // global_addr[31:0]
                     (unsigned int)((ga >> 32) & 0x01FFFFFFu) | (2u << 30) }; // [56:32] | type=2
        i32x8 g1 = { (int)((2u << 16) | (1u << 20) | (5u << 22)), // data_size=4B, pad_en, interval=64dw (amount=1dw)
                     0x00010000,   // [63:48]=tensor_dim0 lo16 = 1
                     0x00400000,   // [95:80]=tensor_dim1 lo16 = 64
                     0x00010000,   // [127:112]=tile_dim0 = 1
                     0x00400040,   // [143:128]=tile_dim1=64, [159:144]=tile_dim2=64
                     128,          // [191:160]=tensor_dim0_stride = 128 elems
                     0x20000000,   // [223:208]=tensor_dim1_stride lo16 = 8192
                     0 };
        i32x4 g2 = { 64, 0, 0, 0 };  // tensor_dim2 = 64
        i32x4 g3 = { 0, 0, 0, 0 };
#if TDM_SIX_ARG
        i32x8 g4 = { 0, 0, 0, 0, 0, 0, 0, 0 };
        __builtin_amdgcn_tensor_load_to_lds(g0, g1, g2, g3, g4, 0);
#else
        __builtin_amdgcn_tensor_load_to_lds(g0, g1, g2, g3, 0);
#endif
        __builtin_amdgcn_s_wait_tensorcnt(0);
    }
#else
    for (int e = tid; e < 64 * 64; e += 128) {
        const int p = e >> 6, q = e & 63;
        sA[p * LDA + q] = Xb[(size_t)(p * 64 + q) * 128];
    }
#endif
    // ---- Init V: identity-ish + deterministic perturbation, zero-padded
    if (tid < 64) {
#pragma unroll
        for (int c = 0; c < 16; ++c)
            sV[tid * LDV + c] =
                (c < NPC) ? ((tid == c ? 1.0f : 0.0f) +
                             0.03125f * (float)((tid * 7 + c * 3) & 7))
                          : 0.0f;
    }
    __syncthreads();

    // ---- cov = A @ A (bpq,bqr->bpr with both operands = A); also store cov^T
    {
        v8f acc[4];
        rowblock64(sA, sA, LDA, row0, lane, acc);
        const int half = lane >> 4;
#pragma unroll
        for (int tn = 0; tn < 4; ++tn) {
            const int n = tn * 16 + (lane & 15);
#pragma unroll
            for (int r = 0; r < 8; ++r) {
                const int mrow = row0 + r + 8 * half;
                sC [mrow * LDA + n] = acc[tn][r];
                sCT[n * LDA + mrow] = acc[tn][r];
            }
        }
    }
    __syncthreads();

    // ---- G = cov @ cov^T (eigvecs of G = left singular vectors of cov);
    //      registers first, then overwrite sC with G.
    {
        v8f g[4];
        rowblock64(sC, sCT, LDA, row0, lane, g);
        __syncthreads();
        const int half = lane >> 4;
#pragma unroll
        for (int tn = 0; tn < 4; ++tn) {
            const int n = tn * 16 + (lane & 15);
#pragma unroll
            for (int r = 0; r < 8; ++r)
                sC[(row0 + r + 8 * half) * LDA + n] = g[tn][r];
        }
    }
    __syncthreads();

    // ---- Blocked power iteration: V <- orth(G @ V) ----
    for (int it = 0; it < NITER; ++it) {
        {   // W = G @ V  (64x64 @ 64x16, one tile per wave)
            v8f d = tile64(sC, sV, LDV, row0, 0, lane);
            const int half = lane >> 4;
            const int n = lane & 15;
#pragma unroll
            for (int r = 0; r < 8; ++r)
                sW[(row0 + r + 8 * half) * LDV + n] = d[r];
        }
        __syncthreads();

        // Modified Gram-Schmidt on 4 columns (rows are thread-local; only the
        // dot products need cross-thread traffic: shfl_xor + 2-way LDS combine)
        for (int j = 0; j < NPC; ++j) {
            for (int i = 0; i < j; ++i) {
                float prod = 0.f;
                if (tid < 64) prod = sW[tid * LDV + i] * sW[tid * LDV + j];
                float r = waveRed(prod);
                __syncthreads();
                if ((tid & 31) == 0) sPart[wave] = r;
                __syncthreads();
                const float s = sPart[0] + sPart[1];
                if (tid < 64) sW[tid * LDV + j] -= s * sW[tid * LDV + i];
            }
            float prod = 0.f;
            if (tid < 64) { const float w = sW[tid * LDV + j]; prod = w * w; }
            float r = waveRed(prod);
            __syncthreads();
            if ((tid & 31) == 0) sPart[wave] = r;
            __syncthreads();
            const float inv = rsqrtf(fmaxf(sPart[0] + sPart[1], 1e-20f));
            if (tid < 64) sW[tid * LDV + j] *= inv;
        }
        __syncthreads();
        // V = [W_orth | 0]
        if (tid < 64) {
#pragma unroll
            for (int c = 0; c < 16; ++c)
                sV[tid * LDV + c] = (c < NPC) ? sW[tid * LDV + c] : 0.0f;
        }
        __syncthreads();
    }

    // ---- proj = A @ V ; reshape(m,-1,n) preserves flat order:
    //      out_flat[b*256 + p*4 + k]
    {
        v8f d = tile64(sA, sV, LDV, row0, 0, lane);
        const int half = lane >> 4;
        const int n = lane & 15;
        if (n < NPC) {
            float* ob = out + (size_t)b * 256;
#pragma unroll
            for (int r = 0; r < 8; ++r)
                ob[(row0 + r + 8 * half) * NPC + n] = d[r];
        }
    }
}

extern "C" void kernel_launch(void* const* d_in, const int* in_sizes, int n_in,
                              void* d_out, int out_size, void* d_ws, size_t ws_size,
                              hipStream_t stream) {
    (void)in_sizes; (void)n_in; (void)d_ws; (void)ws_size; (void)out_size;
    const float* X = (const float*)d_in[0];
    float* out = (float*)d_out;
    // 8192 batches (m*n = 64*128), one workgroup of 4 wave32 per batch
    pca_wmma_kernel<<<8192, 128, 0, stream>>>(X, out);
}